// GCN_2_44659069943892
// MI455X (gfx1250) — compile-verified
//
#include <hip/hip_runtime.h>
#include <hip/hip_bf16.h>

typedef float v2f __attribute__((ext_vector_type(2)));
typedef float v8f __attribute__((ext_vector_type(8)));

// ---------------- normalization ----------------

__global__ void k_init_deg(float* __restrict__ deg, int n) {
    int i = blockIdx.x * blockDim.x + threadIdx.x;
    if (i < n) deg[i] = 1.0f;  // self-loop weight
}

__global__ void k_deg_edges(float* __restrict__ deg, const int* __restrict__ col,
                            const float* __restrict__ ew, int E) {
    int e = blockIdx.x * blockDim.x + threadIdx.x;
    if (e < E) atomicAdd(&deg[col[e]], ew[e]);
}

__global__ void k_dis(float* __restrict__ deg, int n) {
    int i = blockIdx.x * blockDim.x + threadIdx.x;
    if (i < n) {
        float d = deg[i];
        deg[i] = (d > 0.0f) ? rsqrtf(d) : 0.0f;
    }
}

__global__ void k_norm(float* __restrict__ norm, const float* __restrict__ dis,
                       const int* __restrict__ row, const int* __restrict__ col,
                       const float* __restrict__ ew, int E) {
    int e = blockIdx.x * blockDim.x + threadIdx.x;
    if (e < E) norm[e] = dis[row[e]] * ew[e] * dis[col[e]];
}

// ---------------- layer 1: rank-1 product (fi == 1) ----------------
// hpre[n][c] = x[n] * W1[0][c]
__global__ void k_rank1(const float* __restrict__ x, const float* __restrict__ W,
                        float* __restrict__ out, int cnt, int foSh, int foMask) {
    int t = blockIdx.x * blockDim.x + threadIdx.x;
    if (t >= cnt) return;
    out[t] = x[t >> foSh] * W[t & foMask];
}

// ---------------- WMMA GEMM: Out[M,FO] = A[M,FI] @ W[FI,FO] ----------------
// Compile-time FI (mult of 4), FO (mult of 16); M must be a multiple of 16.
// One wave computes one 16x16 tile with V_WMMA_F32_16X16X4_F32.
// A frag (16x4 f32): lanes 0-15 -> M=lane, K=k0,k0+1 ; lanes 16-31 -> M=lane-16, K=k0+2,k0+3
// B frag (4x16 f32): v.x = W[k0+2*half][col], v.y = W[k0+2*half+1][col]
// D (16x16 f32): VGPR r -> (lane<16 ? M=r : M=8+r), N = lane&15

template <int FI, int FO>
__global__ void k_gemm_wmma(const float* __restrict__ A, const float* __restrict__ W,
                            float* __restrict__ Out, int M) {
    constexpr int NT = FO / 16;
    int wave = (blockIdx.x * blockDim.x + threadIdx.x) >> 5;
    int lane = threadIdx.x & 31;
    int mt = wave / NT;
    int nt = wave - mt * NT;
    if (mt * 16 >= M) return;  // wave-uniform exit; EXEC all-ones for WMMA

    int half = lane >> 4;  // 0: lanes 0-15, 1: lanes 16-31
    int l    = lane & 15;
    const float* arow = A + (size_t)(mt * 16 + l) * FI + half * 2;
    int bcol = nt * 16 + l;

    v8f c = {};
#pragma unroll
    for (int k0 = 0; k0 < FI; k0 += 4) {
        v2f a = *(const v2f*)(arow + k0);  // single 64-bit load
        v2f b;
        int kb = k0 + half * 2;
        b.x = W[(size_t)(kb)     * FO + bcol];
        b.y = W[(size_t)(kb + 1) * FO + bcol];
        c = __builtin_amdgcn_wmma_f32_16x16x4_f32(false, a, false, b, (short)0, c,
                                                  false, false);
    }

    float* orow = Out + (size_t)(mt * 16 + half * 8) * FO + bcol;
#pragma unroll
    for (int r = 0; r < 8; ++r) orow[(size_t)r * FO] = c[r];
}

// ---------------- layer aggregation ----------------

// out[n][c] = bias[c] + hpre[n][c] * dis[n]^2   (bias + self-loop contribution)
__global__ void k_init_out(float* __restrict__ out, const float* __restrict__ hpre,
                           const float* __restrict__ dis, const float* __restrict__ bias,
                           int cnt, int foSh, int foMask) {
    int t = blockIdx.x * blockDim.x + threadIdx.x;
    if (t >= cnt) return;
    int n = t >> foSh;
    int c = t & foMask;
    float d = dis[n];
    out[t] = bias[c] + hpre[t] * d * d;
}

// out[col[e]][:] += hpre[row[e]][:] * norm[e]   (float4 groups, coalesced per edge)
__global__ void k_edge_prop(float* __restrict__ out, const float* __restrict__ hpre,
                            const int* __restrict__ row, const int* __restrict__ col,
                            const float* __restrict__ norm, long long total,
                            int fo, int gSh, int gMask) {
    long long t = (long long)blockIdx.x * blockDim.x + threadIdx.x;
    if (t >= total) return;
    int e = (int)(t >> gSh);
    int g = (int)(t & gMask);
    int r = row[e], cc = col[e];
    float nv = norm[e];
    const float4 h4 = *(const float4*)(hpre + (size_t)r * fo + g * 4);
    float* op = out + (size_t)cc * fo + g * 4;
    atomicAdd(op + 0, h4.x * nv);
    atomicAdd(op + 1, h4.y * nv);
    atomicAdd(op + 2, h4.z * nv);
    atomicAdd(op + 3, h4.w * nv);
}

__global__ void k_relu(float* __restrict__ h, int cnt) {
    int t = blockIdx.x * blockDim.x + threadIdx.x;
    if (t < cnt) h[t] = fmaxf(h[t], 0.0f);
}

// ---------------- pooling + head ----------------

__global__ void k_pool_init(float* __restrict__ pool, int cnt) {
    int t = blockIdx.x * blockDim.x + threadIdx.x;
    if (t < cnt) pool[t] = 0.0f;  // valid: features are post-ReLU (>= 0)
}

__global__ void k_pool_max(unsigned* __restrict__ pool, const float* __restrict__ h,
                           const int* __restrict__ batch, int N) {
    int t = blockIdx.x * blockDim.x + threadIdx.x;
    if (t >= N * 16) return;
    int n = t >> 4, c = t & 15;
    unsigned v = __float_as_uint(h[n * 16 + c]);  // >=0 so uint order == float order
    atomicMax(pool + (size_t)batch[n] * 16 + c, v);
}

__global__ void k_mlp(const float* __restrict__ pool, const float* __restrict__ Wl1,
                      const float* __restrict__ bl1, const float* __restrict__ Wl2,
                      const float* __restrict__ bl2, float* __restrict__ out, int G) {
    int g = blockIdx.x * blockDim.x + threadIdx.x;
    if (g >= G) return;
    float h1[16];
#pragma unroll
    for (int j = 0; j < 16; ++j) {
        float acc = bl1[j];
#pragma unroll
        for (int k = 0; k < 16; ++k) acc += pool[g * 16 + k] * Wl1[k * 16 + j];
        h1[j] = fmaxf(acc, 0.0f);
    }
#pragma unroll
    for (int o = 0; o < 2; ++o) {
        float acc = bl2[o];
#pragma unroll
        for (int k = 0; k < 16; ++k) acc += h1[k] * Wl2[k * 2 + o];
        out[g * 2 + o] = acc;
    }
}

// ---------------- launcher ----------------

static inline int gemm_blocks(int M, int fo) {
    int waves = ((M + 15) / 16) * (fo / 16);
    return (waves + 3) / 4;  // 4 waves (128 threads) per block
}

extern "C" void kernel_launch(void* const* d_in, const int* in_sizes, int n_in,
                              void* d_out, int out_size, void* d_ws, size_t ws_size,
                              hipStream_t stream) {
    const float* x     = (const float*)d_in[0];
    const int*   ei    = (const int*)d_in[1];
    const float* ew    = (const float*)d_in[2];
    const int*   batch = (const int*)d_in[3];

    const int N = in_sizes[0];   // 100000 (multiple of 16)
    const int E = in_sizes[2];   // 3200000
    const int G = out_size / 2;  // 500
    const int* row = ei;
    const int* col = ei + E;

    // workspace layout (floats)
    float* ws = (float*)d_ws;
    size_t NPAD = ((size_t)N + 255) & ~(size_t)255;
    size_t EPAD = ((size_t)E + 255) & ~(size_t)255;
    float* dis  = ws;                       // N     (deg -> dis in place)
    float* norm = ws + NPAD;                // E
    float* hA   = norm + EPAD;              // N*64  (layer output / next input)
    float* hB   = hA + (size_t)N * 64;      // N*64  (GEMM output, pre-aggregation)
    float* pool = hB + (size_t)N * 64;      // G*16

    const int T = 256;

    // GCN symmetric normalization
    k_init_deg<<<(N + T - 1) / T, T, 0, stream>>>(dis, N);
    k_deg_edges<<<(E + T - 1) / T, T, 0, stream>>>(dis, col, ew, E);
    k_dis<<<(N + T - 1) / T, T, 0, stream>>>(dis, N);
    k_norm<<<(E + T - 1) / T, T, 0, stream>>>(norm, dis, row, col, ew, E);

    struct L { int fi, fo, foSh, wi, bi; };
    const L layers[6] = {
        {1, 64, 6, 4, 5}, {64, 64, 6, 6, 7}, {64, 64, 6, 8, 9},
        {64, 32, 5, 10, 11}, {32, 32, 5, 12, 13}, {32, 16, 4, 14, 15},
    };

    const float* hin = x;
    for (int li = 0; li < 6; ++li) {
        const L& c = layers[li];
        const float* W = (const float*)d_in[c.wi];
        const float* b = (const float*)d_in[c.bi];
        int cnt = N * c.fo;

        // GEMM: hB = hin @ W
        switch (li) {
        case 0:
            k_rank1<<<(cnt + T - 1) / T, T, 0, stream>>>(x, W, hB, cnt, 6, 63);
            break;
        case 1:
        case 2:
            k_gemm_wmma<64, 64><<<gemm_blocks(N, 64), 128, 0, stream>>>(hin, W, hB, N);
            break;
        case 3:
            k_gemm_wmma<64, 32><<<gemm_blocks(N, 32), 128, 0, stream>>>(hin, W, hB, N);
            break;
        case 4:
            k_gemm_wmma<32, 32><<<gemm_blocks(N, 32), 128, 0, stream>>>(hin, W, hB, N);
            break;
        case 5:
            k_gemm_wmma<32, 16><<<gemm_blocks(N, 16), 128, 0, stream>>>(hin, W, hB, N);
            break;
        }

        // hA = bias + self-loop term
        k_init_out<<<(cnt + T - 1) / T, T, 0, stream>>>(hA, hB, dis, b, cnt, c.foSh,
                                                        c.fo - 1);

        // scatter-add over edges
        int gSh = c.foSh - 2;  // float4 groups per node
        long long total = (long long)E << gSh;
        int blocks = (int)((total + T - 1) / T);
        k_edge_prop<<<blocks, T, 0, stream>>>(hA, hB, row, col, norm, total, c.fo, gSh,
                                              (1 << gSh) - 1);

        k_relu<<<(cnt + T - 1) / T, T, 0, stream>>>(hA, cnt);
        hin = hA;
    }

    // global max pool (post-ReLU, so uint-bit atomicMax with 0-init is exact)
    k_pool_init<<<(G * 16 + T - 1) / T, T, 0, stream>>>(pool, G * 16);
    k_pool_max<<<(N * 16 + T - 1) / T, T, 0, stream>>>((unsigned*)pool, hA, batch, N);

    k_mlp<<<(G + T - 1) / T, T, 0, stream>>>(pool, (const float*)d_in[16],
                                             (const float*)d_in[17],
                                             (const float*)d_in[18],
                                             (const float*)d_in[19], (float*)d_out, G);
}